// GlobalAttentionGraph_41205916237954
// MI455X (gfx1250) — compile-verified
//
#include <hip/hip_runtime.h>
#include <math.h>

typedef float v2f __attribute__((ext_vector_type(2)));
typedef float v8f __attribute__((ext_vector_type(8)));

#define HH 256
#define NB 32
#define NN 1024
#define NROWS (NB * NN)

// Full-precision CDNA5 matrix op: D(16x16,f32) += A(16x4,f32) x B(4x16,f32)
__device__ __forceinline__ v8f wmma_f32_k4(v2f a, v2f b, v8f c) {
  return __builtin_amdgcn_wmma_f32_16x16x4_f32(
      /*neg_a=*/false, a, /*neg_b=*/false, b,
      /*c_mod=*/(short)0, c, /*reuse_a=*/false, /*reuse_b=*/false);
}

// Paired-K helper: one float4 per operand feeds TWO K=4 WMMAs.
// lo covers h in {h0,h0+1,h0+4,h0+5}, hi covers {h0+2,h0+3,h0+6,h0+7}
// (same K-permutation on A and B => valid GEMM over [h0, h0+8)).
__device__ __forceinline__ void wmma_pair(float4 a4, float4 b4, v8f& acc) {
  v2f alo; alo.x = a4.x; alo.y = a4.y;
  v2f ahi; ahi.x = a4.z; ahi.y = a4.w;
  v2f blo; blo.x = b4.x; blo.y = b4.y;
  v2f bhi; bhi.x = b4.z; bhi.y = b4.w;
  acc = wmma_f32_k4(alo, blo, acc);
  acc = wmma_f32_k4(ahi, bhi, acc);
}

// ---------------------------------------------------------------------------
// Y[r, c] = sum_h X[r,h] * W[c,h] + bias[c]      (X: NROWSx256, W: 256x256)
// One wave computes a 16-row x 64-col tile. b128 fragment loads via paired-K.
// ---------------------------------------------------------------------------
__global__ __launch_bounds__(256) void gemm256(
    const float* __restrict__ X, const float* __restrict__ W,
    const float* __restrict__ bias, float* __restrict__ Y) {
  const int lane = threadIdx.x & 31;
  const int wid  = blockIdx.x * 8 + (threadIdx.x >> 5);
  const int rt   = wid >> 2;            // 16-row tile index
  const int cgp  = wid & 3;             // 64-column group
  const int mrow = lane & 15;
  const int hsel = lane >> 4;
  const int hq   = hsel << 2;           // 4*hsel

  const float* xrow = X + (size_t)(rt * 16 + mrow) * HH + hq;
  const float* wc[4];
#pragma unroll
  for (int t = 0; t < 4; ++t)
    wc[t] = W + (size_t)(cgp * 64 + t * 16 + mrow) * HH + hq;

  v8f zero = {0.f, 0.f, 0.f, 0.f, 0.f, 0.f, 0.f, 0.f};
  v8f acc[4];
#pragma unroll
  for (int t = 0; t < 4; ++t) acc[t] = zero;

#pragma unroll 4
  for (int h0 = 0; h0 < HH; h0 += 8) {
    float4 a4 = *(const float4*)(xrow + h0);
#pragma unroll
    for (int t = 0; t < 4; ++t) {
      float4 b4 = *(const float4*)(wc[t] + h0);
      wmma_pair(a4, b4, acc[t]);
    }
  }

#pragma unroll
  for (int t = 0; t < 4; ++t) {
    const int col  = cgp * 64 + t * 16 + mrow;
    const float bv = bias[col];
    float* yp = Y + (size_t)(rt * 16 + 8 * hsel) * HH + col;
#pragma unroll
    for (int v = 0; v < 8; ++v) yp[(size_t)v * HH] = acc[t][v] + bv;
  }
}

// ---------------------------------------------------------------------------
// Flash attention: one wave per (batch, 16-query tile). Online softmax over
// 64 key tiles; O (16x256) kept in 128 accumulator VGPRs. Q tile staged in
// LDS with GLOBAL_LOAD_ASYNC_TO_LDS_B128 (ASYNCcnt), stride 260 padding.
// P converted C/D->A layout through a 20-float-stride LDS bounce buffer
// (16B-aligned ds_load_b128, conflict free).
// ---------------------------------------------------------------------------
__global__ __launch_bounds__(32) void flash_attn(
    const float* __restrict__ Q, const float* __restrict__ K,
    const float* __restrict__ V, const float* __restrict__ masks,
    float* __restrict__ O) {
  __shared__ float qs[16 * 260];
  __shared__ float pbuf[16 * 20];

  const int lane  = threadIdx.x & 31;
  const int b     = blockIdx.x >> 6;
  const int qt    = blockIdx.x & 63;
  const int q0    = qt << 4;
  const size_t base = (size_t)b * NN * HH;
  const int mrow  = lane & 15;
  const int hsel  = lane >> 4;
  const int hq    = hsel << 2;

  // Async-stage Q tile (16x256 f32 = 16KB) into LDS: 32 x (32 lanes x 16B).
#pragma unroll 4
  for (int i = 0; i < 32; ++i) {
    const int c  = i * 32 + lane;       // 16-byte chunk id, 64 per row
    const int rr = c >> 6;
    const int cf = (c & 63) << 2;
    const float* gp = Q + base + (size_t)(q0 + rr) * HH + cf;
    const unsigned lp = (unsigned)(size_t)(const void*)(qs + rr * 260 + cf);
    asm volatile("global_load_async_to_lds_b128 %0, %1, off"
                 :
                 : "v"(lp), "v"(gp)
                 : "memory");
  }
  asm volatile("s_wait_asynccnt 0" ::: "memory");
  __syncthreads();

  float mi[8], mst[8], lst[8];
#pragma unroll
  for (int v = 0; v < 8; ++v) {
    mi[v]  = masks[b * NN + q0 + v + 8 * hsel];
    mst[v] = -3.0e38f;
    lst[v] = 0.0f;
  }
  v8f zero = {0.f, 0.f, 0.f, 0.f, 0.f, 0.f, 0.f, 0.f};
  v8f o[16];
#pragma unroll
  for (int t = 0; t < 16; ++t) o[t] = zero;

  const float* qlds = qs + mrow * 260 + hq;

  for (int kt = 0; kt < 64; ++kt) {
    const float mj = masks[b * NN + kt * 16 + mrow];
    const float* krow = K + base + (size_t)(kt * 16 + mrow) * HH + hq;
    if (kt < 63) __builtin_prefetch(krow + 16 * HH, 0, 0);

    // S = Q_tile x K_tile^T  (16x16, K=256 as 32 paired b128 steps)
    v8f s = zero;
#pragma unroll 4
    for (int h0 = 0; h0 < HH; h0 += 8) {
      float4 a4 = *(const float4*)(qlds + h0);
      float4 b4 = *(const float4*)(krow + h0);
      wmma_pair(a4, b4, s);
    }

    // scale + additive mask bias, then online softmax update (rows live in
    // 16-lane halves of each C/D VGPR -> width-16 shuffle reductions)
#pragma unroll
    for (int v = 0; v < 8; ++v) {
      const float mm = mi[v] * mj;
      float sv = s[v] * 0.0625f + (mm - 100000.0f * (1.0f - mm));
      float rm = sv;
      rm = fmaxf(rm, __shfl_xor(rm, 1, 16));
      rm = fmaxf(rm, __shfl_xor(rm, 2, 16));
      rm = fmaxf(rm, __shfl_xor(rm, 4, 16));
      rm = fmaxf(rm, __shfl_xor(rm, 8, 16));
      const float mnew  = fmaxf(mst[v], rm);
      const float alpha = __expf(mst[v] - mnew);
      const float p     = __expf(sv - mnew);
      float rs = p;
      rs += __shfl_xor(rs, 1, 16);
      rs += __shfl_xor(rs, 2, 16);
      rs += __shfl_xor(rs, 4, 16);
      rs += __shfl_xor(rs, 8, 16);
      lst[v] = lst[v] * alpha + rs;
      mst[v] = mnew;
      pbuf[(v + 8 * hsel) * 20 + mrow] = p;   // C/D layout -> LDS
#pragma unroll
      for (int t = 0; t < 16; ++t) o[t][v] *= alpha;
    }
    asm volatile("s_wait_dscnt 0" ::: "memory");

    // O += P x V_tile (K=16 as 2 paired-K spans of 8 keys, 16 column tiles)
#pragma unroll
    for (int sp = 0; sp < 2; ++sp) {
      float4 p4 = *(const float4*)(pbuf + mrow * 20 + sp * 8 + hq);
      v2f plo; plo.x = p4.x; plo.y = p4.y;   // keys sp*8 + 4*hsel + {0,1}
      v2f phi; phi.x = p4.z; phi.y = p4.w;   // keys sp*8 + 4*hsel + {2,3}
      const float* vblo =
          V + base + (size_t)(kt * 16 + sp * 8 + hq) * HH + mrow;
      const float* vbhi = vblo + 2 * HH;
#pragma unroll
      for (int t = 0; t < 16; ++t) {
        v2f b1; b1.x = vblo[t * 16]; b1.y = vblo[HH + t * 16];
        o[t] = wmma_f32_k4(plo, b1, o[t]);
        v2f b2; b2.x = vbhi[t * 16]; b2.y = vbhi[HH + t * 16];
        o[t] = wmma_f32_k4(phi, b2, o[t]);
      }
    }
  }

#pragma unroll
  for (int v = 0; v < 8; ++v) lst[v] = 1.0f / lst[v];
#pragma unroll
  for (int t = 0; t < 16; ++t) {
    float* op = O + base + (size_t)(q0 + 8 * hsel) * HH + t * 16 + mrow;
#pragma unroll
    for (int v = 0; v < 8; ++v) op[(size_t)v * HH] = o[t][v] * lst[v];
  }
}

// ---------------------------------------------------------------------------
// Row-wise LayerNorm (+optional ReLU). One wave per 256-wide row.
// ---------------------------------------------------------------------------
__global__ __launch_bounds__(256) void ln_rows(
    const float* X, float* Y, const float* __restrict__ g,
    const float* __restrict__ be, int relu) {
  const int lane = threadIdx.x & 31;
  const int row  = blockIdx.x * 8 + (threadIdx.x >> 5);
  const float* xr = X + (size_t)row * HH + lane * 8;
  float4 x0 = *(const float4*)(xr);
  float4 x1 = *(const float4*)(xr + 4);
  float x[8] = {x0.x, x0.y, x0.z, x0.w, x1.x, x1.y, x1.z, x1.w};
  float s = 0.f, q = 0.f;
#pragma unroll
  for (int e = 0; e < 8; ++e) { s += x[e]; q += x[e] * x[e]; }
#pragma unroll
  for (int d = 1; d < 32; d <<= 1) {
    s += __shfl_xor(s, d, 32);
    q += __shfl_xor(q, d, 32);
  }
  const float mu   = s * (1.0f / HH);
  const float var  = q * (1.0f / HH) - mu * mu;
  const float rstd = rsqrtf(var + 1e-5f);
  float* yr = Y + (size_t)row * HH + lane * 8;
#pragma unroll
  for (int e = 0; e < 8; ++e) {
    float vv = (x[e] - mu) * rstd * g[lane * 8 + e] + be[lane * 8 + e];
    if (relu) vv = fmaxf(vv, 0.0f);
    yr[e] = vv;
  }
}

// ---------------------------------------------------------------------------
// out = LN(T; gT,bT) + LN(R; gR,bR)   (+optional ReLU). One wave per row.
// ---------------------------------------------------------------------------
__global__ __launch_bounds__(256) void combine_rows(
    const float* __restrict__ T, const float* __restrict__ R,
    const float* __restrict__ gT, const float* __restrict__ bT,
    const float* __restrict__ gR, const float* __restrict__ bR,
    float* __restrict__ Y, int relu) {
  const int lane = threadIdx.x & 31;
  const int row  = blockIdx.x * 8 + (threadIdx.x >> 5);
  const float* tr = T + (size_t)row * HH + lane * 8;
  const float* rr = R + (size_t)row * HH + lane * 8;
  float4 t0 = *(const float4*)(tr);
  float4 t1 = *(const float4*)(tr + 4);
  float4 r0 = *(const float4*)(rr);
  float4 r1 = *(const float4*)(rr + 4);
  float tv[8] = {t0.x, t0.y, t0.z, t0.w, t1.x, t1.y, t1.z, t1.w};
  float rv[8] = {r0.x, r0.y, r0.z, r0.w, r1.x, r1.y, r1.z, r1.w};
  float st = 0.f, qt = 0.f, sr = 0.f, qr = 0.f;
#pragma unroll
  for (int e = 0; e < 8; ++e) {
    st += tv[e]; qt += tv[e] * tv[e];
    sr += rv[e]; qr += rv[e] * rv[e];
  }
#pragma unroll
  for (int d = 1; d < 32; d <<= 1) {
    st += __shfl_xor(st, d, 32);
    qt += __shfl_xor(qt, d, 32);
    sr += __shfl_xor(sr, d, 32);
    qr += __shfl_xor(qr, d, 32);
  }
  const float muT = st * (1.0f / HH);
  const float rsT = rsqrtf(qt * (1.0f / HH) - muT * muT + 1e-5f);
  const float muR = sr * (1.0f / HH);
  const float rsR = rsqrtf(qr * (1.0f / HH) - muR * muR + 1e-5f);
  float* yr = Y + (size_t)row * HH + lane * 8;
#pragma unroll
  for (int e = 0; e < 8; ++e) {
    const int c = lane * 8 + e;
    float vv = (tv[e] - muT) * rsT * gT[c] + bT[c] +
               (rv[e] - muR) * rsR * gR[c] + bR[c];
    if (relu) vv = fmaxf(vv, 0.0f);
    yr[e] = vv;
  }
}

extern "C" void kernel_launch(void* const* d_in, const int* in_sizes, int n_in,
                              void* d_out, int out_size, void* d_ws,
                              size_t ws_size, hipStream_t stream) {
  const float* feats = (const float*)d_in[0];
  const float* masks = (const float*)d_in[1];
  const float* qw  = (const float*)d_in[2];
  const float* qb  = (const float*)d_in[3];
  const float* kw  = (const float*)d_in[4];
  const float* kb  = (const float*)d_in[5];
  const float* vw  = (const float*)d_in[6];
  const float* vb  = (const float*)d_in[7];
  const float* ag  = (const float*)d_in[8];
  const float* ab  = (const float*)d_in[9];
  const float* cw  = (const float*)d_in[10];
  const float* cb  = (const float*)d_in[11];
  const float* cg  = (const float*)d_in[12];
  const float* cbl = (const float*)d_in[13];
  const float* lw  = (const float*)d_in[14];
  const float* lb  = (const float*)d_in[15];
  const float* lg  = (const float*)d_in[16];
  const float* lbl = (const float*)d_in[17];

  const size_t E = (size_t)NROWS * HH;
  float* qbuf = (float*)d_ws;      // also reused as ca-projection output
  float* kbuf = qbuf + E;
  float* vbuf = kbuf + E;
  float* rbuf = vbuf + E;          // residual projection
  float* cbuf = rbuf + E;          // attention context
  float* fbuf = cbuf + E;          // next-layer feats

  for (int i = 0; i < 3; ++i) {
    const float* src = (i == 0) ? feats : fbuf;
    float* dst = (i == 2) ? (float*)d_out : fbuf;
    const size_t wo = (size_t)i * HH * HH;
    const size_t bo = (size_t)i * HH;

    gemm256<<<dim3(1024), dim3(256), 0, stream>>>(src, qw + wo, qb + bo, qbuf);
    gemm256<<<dim3(1024), dim3(256), 0, stream>>>(src, kw + wo, kb + bo, kbuf);
    gemm256<<<dim3(1024), dim3(256), 0, stream>>>(src, vw + wo, vb + bo, vbuf);
    gemm256<<<dim3(1024), dim3(256), 0, stream>>>(src, lw + wo, lb + bo, rbuf);

    flash_attn<<<dim3(2048), dim3(32), 0, stream>>>(qbuf, kbuf, vbuf, masks,
                                                    cbuf);
    ln_rows<<<dim3(4096), dim3(256), 0, stream>>>(cbuf, cbuf, ag + bo, ab + bo,
                                                  1);
    gemm256<<<dim3(1024), dim3(256), 0, stream>>>(cbuf, cw + wo, cb + bo, qbuf);
    combine_rows<<<dim3(4096), dim3(256), 0, stream>>>(
        qbuf, rbuf, cg + bo, cbl + bo, lg + bo, lbl + bo, dst,
        (i < 2) ? 1 : 0);
  }
  (void)in_sizes; (void)n_in; (void)out_size; (void)ws_size;
}